// QuantumHybridQLSTMTagger_65481071398377
// MI455X (gfx1250) — compile-verified
//
#include <hip/hip_runtime.h>
#include <hip/hip_bf16.h>

typedef __attribute__((ext_vector_type(2))) float v2f;
typedef __attribute__((ext_vector_type(8))) float v8f;

#define TT 128
#define BB 512
#define EE 256
#define HH 16
#define TAGS 64
#define LDS_STRIDE 20   // padded row stride (floats); 8B-aligned pairs, spreads banks

#define ZPRE_BYTES (TT * BB * 64 * 4)          // 16.78 MB
#define WPACK_ELTS (4 * 64 * 32)               // 8192 float2 = 64 KB

__device__ __forceinline__ v8f wmma_f32(v2f a, v2f b, v8f c) {
    return __builtin_amdgcn_wmma_f32_16x16x4_f32(false, a, false, b, (short)0, c,
                                                 false, false);
}

__device__ __forceinline__ float fast_sigmoid(float x) {
    return 1.0f / (1.0f + __expf(-x));
}
__device__ __forceinline__ float fast_tanh(float x) {
    float e = __expf(2.0f * x);
    return (e - 1.0f) / (e + 1.0f);
}

// ---------------------------------------------------------------------------
// Kernel 0: one-time repack of the input-half gate weights (rows 0..255 of
// Wf/Wi/Wg/Wo) into WMMA B-fragment order so the hot GEMM loop issues a
// single coalesced global_load_b64 per (gate, k-chunk):
//   Wpack[((g*64 + kc)*32 + lane)] = { W[kb*16+lo], W[(kb+1)*16+lo] },
//   kb = kc*4 + 2*(lane>>4), lo = lane&15.
// ---------------------------------------------------------------------------
__global__ __launch_bounds__(256) void qlstm_pack_b(
    const float* __restrict__ Wf, const float* __restrict__ Wi,
    const float* __restrict__ Wg, const float* __restrict__ Wo,
    v2f* __restrict__ wpack)
{
    const int idx = blockIdx.x * blockDim.x + threadIdx.x;   // 0..8191
    if (idx >= WPACK_ELTS) return;
    const int lane = idx & 31;
    const int kc   = (idx >> 5) & 63;
    const int g    = idx >> 11;
    const int lo   = lane & 15;
    const int hi   = lane >> 4;
    const float* W = (g == 0) ? Wf : (g == 1) ? Wi : (g == 2) ? Wg : Wo;
    const int kb = kc * 4 + 2 * hi;
    v2f b;
    b.x = W[kb * HH + lo];
    b.y = W[(kb + 1) * HH + lo];
    wpack[idx] = b;
}

// ---------------------------------------------------------------------------
// Kernel 1: fused embedding-gather + input-half gate GEMM (bias folded into
// the WMMA accumulator init).  M = T*B = 65536, N = 64 (4 gates), K = 256.
// One wave per 16-row tile; exact fp32 via chained V_WMMA_F32_16X16X4_F32.
// A frags: one b64 gather-load per lane per k-chunk from the (L2-resident,
// 32.8 MB) embedding table.  B frags: one coalesced b64 load from Wpack.
// ---------------------------------------------------------------------------
__global__ __launch_bounds__(256) void qlstm_pre_gemm(
    const int*   __restrict__ sentence,                       // [T*B]
    const float* __restrict__ emb,                            // [V, 256]
    const v2f*   __restrict__ wpack,                          // [4][64][32]
    const float* __restrict__ bf, const float* __restrict__ bi,
    const float* __restrict__ bg, const float* __restrict__ bo,
    float*       __restrict__ zpre)                           // [T*B, 64]
{
    const int lane = threadIdx.x & 31;
    const int lo   = lane & 15;
    const int hi   = lane >> 4;
    const int wave = blockIdx.x * (blockDim.x >> 5) + (threadIdx.x >> 5);
    const int m0   = wave * 16;

    // gather: lane's M-row token id (lanes 16..31 mirror 0..15)
    const int row = sentence[m0 + lo];
    const float* arow = emb + row * EE;

    const float* bv[4] = {bf, bi, bg, bo};
    v8f acc[4];
#pragma unroll
    for (int g = 0; g < 4; ++g) {
        float b = bv[g][lo];
#pragma unroll
        for (int r = 0; r < 8; ++r) acc[g][r] = b;
    }

#pragma unroll 4
    for (int kc = 0; kc < 64; ++kc) {
        // A frag: 16x4 fp32, lane=M; v0=K(2*hi), v1=K(2*hi+1) -> one b64 load
        v2f a = *(const v2f*)(arow + kc * 4 + 2 * hi);
#pragma unroll
        for (int g = 0; g < 4; ++g)
            acc[g] = wmma_f32(a, wpack[(g * 64 + kc) * 32 + lane], acc[g]);
    }

#pragma unroll
    for (int g = 0; g < 4; ++g)
#pragma unroll
        for (int r = 0; r < 8; ++r)
            zpre[(m0 + r + 8 * hi) * 64 + g * 16 + lo] = acc[g][r];
}

// ---------------------------------------------------------------------------
// Kernel 2: the recurrent scan.  LSTM rows are independent per batch sample,
// so each wave owns a 16-row batch tile and runs all 128 steps privately
// (zero inter-wave sync).  Per step:
//   z = Zpre[t] + h @ Wh        (4 gates, fp32 WMMA, K=16)
//   gate = cumprod(cos(z+th))   (width-16 shfl_up lane scan)
//   c,h update; h -> LDS -> A frags (C-layout -> A-layout transpose)
//   logits = h @ Wt + bt        (fp32 WMMA) + fused log-softmax over 64 tags
// Zpre for step t+1 is prefetched into registers at the top of step t so the
// serial loop's L2 latency hides behind the transcendental/WMMA work.
// ---------------------------------------------------------------------------
__global__ __launch_bounds__(32) void qlstm_scan(
    const float* __restrict__ zpre,                           // [T*B, 64]
    const float* __restrict__ Wf, const float* __restrict__ Wi,
    const float* __restrict__ Wg, const float* __restrict__ Wo,
    const float* __restrict__ thf, const float* __restrict__ thi,
    const float* __restrict__ thg, const float* __restrict__ tho,
    const float* __restrict__ Wt,  const float* __restrict__ bt,
    float*       __restrict__ out)                            // [T*B, 64]
{
    __shared__ float hlds[16 * LDS_STRIDE];

    const int lane = threadIdx.x & 31;
    const int lo   = lane & 15;
    const int hi   = lane >> 4;
    const int b0   = blockIdx.x * 16;   // batch tile

    // Recurrent-weight B fragments: rows 256..271 of each W*, K=16 (4 chunks).
    const float* Wh[4] = {Wf, Wi, Wg, Wo};
    v2f bW[4][4];
#pragma unroll
    for (int g = 0; g < 4; ++g)
#pragma unroll
        for (int kc = 0; kc < 4; ++kc) {
            const int kb = kc * 4 + 2 * hi;
            bW[g][kc].x = Wh[g][(EE + kb) * HH + lo];
            bW[g][kc].y = Wh[g][(EE + kb + 1) * HH + lo];
        }
    // Tagger B fragments: Wt is [16, 64].
    v2f bT[4][4];
#pragma unroll
    for (int j = 0; j < 4; ++j)
#pragma unroll
        for (int kc = 0; kc < 4; ++kc) {
            const int kb = kc * 4 + 2 * hi;
            bT[j][kc].x = Wt[kb * TAGS + j * 16 + lo];
            bT[j][kc].y = Wt[(kb + 1) * TAGS + j * 16 + lo];
        }

    const float th[4] = {thf[lo], thi[lo], thg[lo], tho[lo]};
    float btv[4];
#pragma unroll
    for (int j = 0; j < 4; ++j) btv[j] = bt[j * 16 + lo];

    v8f c = {};          // cell state, C-layout
    v2f a[4];            // A frags of h_{t-1}; h_0 = 0
#pragma unroll
    for (int kc = 0; kc < 4; ++kc) { a[kc].x = 0.0f; a[kc].y = 0.0f; }

    // software pipeline: zn holds Zpre for the *next* consumed step
    v8f zn[4];
    {
        const int base0 = (0 * BB + b0) * 64;
#pragma unroll
        for (int g = 0; g < 4; ++g)
#pragma unroll
            for (int r = 0; r < 8; ++r)
                zn[g][r] = zpre[base0 + (r + 8 * hi) * 64 + g * 16 + lo];
    }

    for (int t = 0; t < TT; ++t) {
        const int base = (t * BB + b0) * 64;

        // rotate pipeline: consume zn, immediately issue loads for t+1
        v8f z[4];
#pragma unroll
        for (int g = 0; g < 4; ++g) z[g] = zn[g];
        {
            const int tn    = (t + 1 < TT) ? (t + 1) : t;  // harmless re-load on last
            const int basen = (tn * BB + b0) * 64;
#pragma unroll
            for (int g = 0; g < 4; ++g)
#pragma unroll
                for (int r = 0; r < 8; ++r)
                    zn[g][r] = zpre[basen + (r + 8 * hi) * 64 + g * 16 + lo];
        }

        // z_g += h_{t-1} @ Wh_g
#pragma unroll
        for (int g = 0; g < 4; ++g)
#pragma unroll
            for (int kc = 0; kc < 4; ++kc)
                z[g] = wmma_f32(a[kc], bW[g][kc], z[g]);

        // quantum gate: cumprod(cos(z+theta)) over the 16 wires (lane scan)
#pragma unroll
        for (int g = 0; g < 4; ++g)
#pragma unroll
            for (int r = 0; r < 8; ++r) {
                float u = __cosf(z[g][r] + th[g]);
#pragma unroll
                for (int d = 1; d < 16; d <<= 1) {
                    float up = __shfl_up(u, (unsigned)d, 16);
                    if (lo >= d) u *= up;
                }
                z[g][r] = u;
            }

        // LSTM elementwise update
        v8f hreg;
#pragma unroll
        for (int r = 0; r < 8; ++r) {
            float f  = fast_sigmoid(z[0][r]);
            float ii = fast_sigmoid(z[1][r]);
            float gg = fast_tanh(z[2][r]);
            float oo = fast_sigmoid(z[3][r]);
            float cn = f * c[r] + ii * gg;
            c[r]     = cn;
            hreg[r]  = oo * fast_tanh(cn);
        }

        // C-layout -> A-layout transpose of h through LDS (single-wave WG:
        // barrier degenerates; only ds-count waits remain)
#pragma unroll
        for (int r = 0; r < 8; ++r)
            hlds[(r + 8 * hi) * LDS_STRIDE + lo] = hreg[r];
        __syncthreads();
#pragma unroll
        for (int kc = 0; kc < 4; ++kc)
            a[kc] = *(const v2f*)&hlds[lo * LDS_STRIDE + kc * 4 + 2 * hi];
        __syncthreads();

        // tagger logits: h_t @ Wt + bt
        v8f L[4];
#pragma unroll
        for (int j = 0; j < 4; ++j)
#pragma unroll
            for (int r = 0; r < 8; ++r) L[j][r] = btv[j];
#pragma unroll
        for (int j = 0; j < 4; ++j)
#pragma unroll
            for (int kc = 0; kc < 4; ++kc)
                L[j] = wmma_f32(a[kc], bT[j][kc], L[j]);

        // fused log-softmax over 64 tags (4 frags x 16 lanes per row)
#pragma unroll
        for (int r = 0; r < 8; ++r) {
            float m = fmaxf(fmaxf(L[0][r], L[1][r]), fmaxf(L[2][r], L[3][r]));
#pragma unroll
            for (int d = 1; d < 16; d <<= 1)
                m = fmaxf(m, __shfl_xor(m, d, 16));
            float s = __expf(L[0][r] - m) + __expf(L[1][r] - m) +
                      __expf(L[2][r] - m) + __expf(L[3][r] - m);
#pragma unroll
            for (int d = 1; d < 16; d <<= 1)
                s += __shfl_xor(s, d, 16);
            const float ls = m + __logf(s);
            const int ob = base + (r + 8 * hi) * 64;
#pragma unroll
            for (int j = 0; j < 4; ++j)
                out[ob + j * 16 + lo] = L[j][r] - ls;
        }
    }
}

extern "C" void kernel_launch(void* const* d_in, const int* in_sizes, int n_in,
                              void* d_out, int out_size, void* d_ws, size_t ws_size,
                              hipStream_t stream) {
    const int*   sentence = (const int*)  d_in[0];
    const float* emb      = (const float*)d_in[1];
    const float* Wf = (const float*)d_in[2];  const float* bf = (const float*)d_in[3];
    const float* Wi = (const float*)d_in[4];  const float* bi = (const float*)d_in[5];
    const float* Wg = (const float*)d_in[6];  const float* bg = (const float*)d_in[7];
    const float* Wo = (const float*)d_in[8];  const float* bo = (const float*)d_in[9];
    const float* thf = (const float*)d_in[10];
    const float* thi = (const float*)d_in[11];
    const float* thg = (const float*)d_in[12];
    const float* tho = (const float*)d_in[13];
    const float* Wt  = (const float*)d_in[14];
    const float* bt  = (const float*)d_in[15];
    float* outp = (float*)d_out;

    float* zpre  = (float*)d_ws;                              // 16.78 MB
    v2f*   wpack = (v2f*)((char*)d_ws + ZPRE_BYTES);          // +64 KB

    // Kernel 0: repack gate weights into B-fragment order (8192 float2).
    qlstm_pack_b<<<(WPACK_ELTS + 255) / 256, 256, 0, stream>>>(Wf, Wi, Wg, Wo,
                                                               wpack);
    // Kernel 1: 65536/16 = 4096 wave-tiles, 8 waves (256 thr) per block.
    qlstm_pre_gemm<<<512, 256, 0, stream>>>(sentence, emb, wpack,
                                            bf, bi, bg, bo, zpre);
    // Kernel 2: 512/16 = 32 independent batch tiles, one wave each.
    qlstm_scan<<<32, 32, 0, stream>>>(zpre, Wf, Wi, Wg, Wo,
                                      thf, thi, thg, tho, Wt, bt, outp);
}